// MLP_9380208575071
// MI455X (gfx1250) — compile-verified
//
#include <hip/hip_runtime.h>
#include <math.h>

// ---------------- problem constants ----------------
#define BB     4
#define NN     1024
#define MM     100
#define NBF    25          // bins per type
#define FEATN  50
#define FEATP  64          // padded feature dim (K for layer0)
#define HP     128         // padded hidden dim (100 -> 128)
#define PI_F   3.1415926f
#define RMAXF  6.0f
#define NATOM  (BB*NN)     // 4096

typedef float v8f __attribute__((ext_vector_type(8)));
typedef float v2f __attribute__((ext_vector_type(2)));

// fp32 WMMA: D(16x16) = A(16x4) * B(4x16) + C   (exact fp32, matches reference)
__device__ __forceinline__ v8f wmma4(v2f a, v2f b, v8f c) {
  return __builtin_amdgcn_wmma_f32_16x16x4_f32(
      /*neg_a=*/false, a, /*neg_b=*/false, b,
      /*c_mod=*/(short)0, c, /*reuse_a=*/false, /*reuse_b=*/false);
}

// ---------------- weight padding ----------------
// dst[t][kp][np] = (kp<K && np<N) ? src[t][kp][np] : 0
__global__ void pad_w(const float* __restrict__ src, float* __restrict__ dst,
                      int T, int K, int N, int KP, int NP) {
  int idx = blockIdx.x * blockDim.x + threadIdx.x;
  int tot = T * KP * NP;
  if (idx >= tot) return;
  int np = idx % NP;
  int r  = idx / NP;
  int kp = r % KP;
  int t  = r / KP;
  float v = 0.f;
  if (kp < K && np < N) v = src[(t * K + kp) * N + np];
  dst[idx] = v;
}

// ---------------- features ----------------
// feat[g][0:64]: 25 bins for neighbor-type 1, 25 for type 8, 14 zero pad
__global__ void feat_kernel(const int* __restrict__ itype,
                            const int* __restrict__ nbr,
                            const float* __restrict__ dR,
                            float* __restrict__ feat) {
  __shared__ float sacc[FEATP];
  const int g = blockIdx.x;          // atom index 0..4095
  const int b = g >> 10;
  const int tid = threadIdx.x;
  if (tid < FEATP) sacc[tid] = 0.f;
  __syncthreads();
  if (tid < MM) {
    const int idx = g * MM + tid;
    const int j = nbr[idx];
    const float r = dR[idx * 4 + 0];
    const int nt = (j == 0) ? 0 : itype[b * NN + (j - 1)];
    const int off = (nt == 1) ? 0 : (nt == 8) ? NBF : -1;
    if (off >= 0 && r > 0.f && r < RMAXF) {
      const float win = 0.5f * cosf(PI_F * r) + 0.5f;
      #pragma unroll
      for (int i = 0; i < NBF; ++i) {
        float d = r - (0.5f + (float)i * (5.5f / 24.f));
        atomicAdd(&sacc[off + i], expf(-d * d) * win);
      }
    }
  }
  __syncthreads();
  if (tid < FEATP) feat[g * FEATP + tid] = sacc[tid];
}

// ---------------- zero output ----------------
__global__ void zero_out(float* __restrict__ out, int n) {
  int i = blockIdx.x * blockDim.x + threadIdx.x;
  if (i < n) out[i] = 0.f;
}

// ---------------- WMMA GEMM helpers ----------------
// forward: H[16][128] = tanh(X[16][KP] @ W[KP][128] + bias)
__device__ __forceinline__ void gemm_fwd_tanh(const float* x, int xs,
                                              const float* __restrict__ W,
                                              const float* __restrict__ bias,
                                              float* h, int KP,
                                              int lane, int wave) {
  const int mrow = lane & 15;            // A row / B,C,D column (per ISA layout)
  const int half = (lane >> 4) & 1;      // K-half select / M-half select
  const int kh   = half * 2;
  for (int nt = wave; nt < 8; nt += 4) {
    v8f acc = {0.f,0.f,0.f,0.f,0.f,0.f,0.f,0.f};
    const int ncol = nt * 16 + mrow;
    for (int kt = 0; kt < (KP >> 2); ++kt) {
      const int k0 = (kt << 2) + kh;
      v2f a, bb;
      a[0]  = x[mrow * xs + k0];
      a[1]  = x[mrow * xs + k0 + 1];
      bb[0] = W[k0 * HP + ncol];
      bb[1] = W[(k0 + 1) * HP + ncol];
      acc = wmma4(a, bb, acc);
    }
    const float bi = bias[ncol];
    #pragma unroll
    for (int r = 0; r < 8; ++r) {
      h[(r + half * 8) * HP + ncol] = tanhf(acc[r] + bi);
    }
  }
}

// backward: OUT[16][ntiles*16] = G[16][128] @ W^T  (W is [KOUT][128] row-major)
__device__ __forceinline__ void gemm_bT(const float* g,
                                        const float* __restrict__ W,
                                        float* out, int ntiles,
                                        int lane, int wave) {
  const int mrow = lane & 15;
  const int half = (lane >> 4) & 1;
  const int kh   = half * 2;
  for (int nt = wave; nt < ntiles; nt += 4) {
    v8f acc = {0.f,0.f,0.f,0.f,0.f,0.f,0.f,0.f};
    const int col = nt * 16 + mrow;      // output column = row index of W
    for (int kt = 0; kt < 32; ++kt) {    // contraction over 128
      const int k0 = (kt << 2) + kh;
      v2f a, bb;
      a[0]  = g[mrow * HP + k0];
      a[1]  = g[mrow * HP + k0 + 1];
      bb[0] = W[col * HP + k0];          // W^T[k][col] = W[col][k] (contiguous)
      bb[1] = W[col * HP + k0 + 1];
      acc = wmma4(a, bb, acc);
    }
    #pragma unroll
    for (int r = 0; r < 8; ++r) {
      out[(r + half * 8) * HP + col] = acc[r];
    }
  }
}

// ---------------- fused MLP fwd+bwd per 16-atom tile ----------------
__global__ void __launch_bounds__(128)
mlp_kernel(const int* __restrict__ itype, const float* __restrict__ feat,
           const float* __restrict__ W0p, const float* __restrict__ W1p,
           const float* __restrict__ W2p, const float* __restrict__ W3p,
           const float* __restrict__ b0p, const float* __restrict__ b1p,
           const float* __restrict__ b2p, const float* __restrict__ b3p,
           float* __restrict__ dE, float* __restrict__ out) {
  __shared__ float sX [16 * HP];    // activations, then dx in backward
  __shared__ float sH0[16 * HP];
  __shared__ float sH1[16 * HP];    // also gemm2/gemm3 output buffer
  __shared__ float sH2[16 * HP];    // also gemm1 output buffer
  __shared__ float sGZ[16 * HP];
  __shared__ float sX0[16 * FEATP];
  __shared__ float sW3c[HP];
  __shared__ float sEi[16];
  __shared__ int   sTy[16];

  const int tid  = threadIdx.x;
  const int lane = tid & 31;
  const int wave = tid >> 5;
  const int base = blockIdx.x * 16;

  for (int e = tid; e < 16 * FEATP; e += 128) sX0[e] = feat[base * FEATP + e];
  if (tid < 16) sTy[tid] = itype[base + tid];
  __syncthreads();

  for (int t = 0; t < 2; ++t) {
    const float* W0 = W0p + t * FEATP * HP;
    const float* W1 = W1p + t * HP * HP;
    const float* W2 = W2p + t * HP * HP;
    const float* W3 = W3p + t * HP * 16;
    const float* b0 = b0p + t * HP;
    const float* b1 = b1p + t * HP;
    const float* b2 = b2p + t * HP;
    const float  b3 = b3p[t * 16];

    if (tid < HP) sW3c[tid] = W3[tid * 16];   // dEi/dx3 column
    __syncthreads();

    // ---- forward ----
    gemm_fwd_tanh(sX0, FEATP, W0, b0, sH0, FEATP, lane, wave);
    __syncthreads();
    for (int e = tid; e < 16 * HP; e += 128) {  // x1 = h0 + concat(x0,x0)
      int a = e >> 7, j = e & 127;
      float sk = (j < FEATN) ? sX0[a * FEATP + j]
               : (j < 2 * FEATN) ? sX0[a * FEATP + j - FEATN] : 0.f;
      sX[e] = sH0[e] + sk;
    }
    __syncthreads();
    gemm_fwd_tanh(sX, HP, W1, b1, sH1, HP, lane, wave);
    __syncthreads();
    for (int e = tid; e < 16 * HP; e += 128) sX[e] += sH1[e];   // x2
    __syncthreads();
    gemm_fwd_tanh(sX, HP, W2, b2, sH2, HP, lane, wave);
    __syncthreads();
    for (int e = tid; e < 16 * HP; e += 128) sX[e] += sH2[e];   // x3
    __syncthreads();
    // head: Ei = x3 @ W3[:,0] + b3  (one WMMA ntile, wave 0)
    if (wave == 0) {
      const int mrow = lane & 15;
      const int half = (lane >> 4) & 1;
      const int kh   = half * 2;
      v8f acc = {0.f,0.f,0.f,0.f,0.f,0.f,0.f,0.f};
      for (int kt = 0; kt < 32; ++kt) {
        const int k0 = (kt << 2) + kh;
        v2f a, bb;
        a[0]  = sX[mrow * HP + k0];
        a[1]  = sX[mrow * HP + k0 + 1];
        bb[0] = W3[k0 * 16 + mrow];
        bb[1] = W3[(k0 + 1) * 16 + mrow];
        acc = wmma4(a, bb, acc);
      }
      if ((lane & 15) == 0) {
        #pragma unroll
        for (int r = 0; r < 8; ++r) sEi[r + half * 8] = acc[r] + b3;
      }
    }
    __syncthreads();

    // ---- backward:  dE/dfeat ----
    for (int e = tid; e < 16 * HP; e += 128) {      // dz2 = W3col * (1-h2^2)
      int j = e & 127;
      sGZ[e] = sW3c[j] * (1.f - sH2[e] * sH2[e]);
    }
    __syncthreads();
    gemm_bT(sGZ, W2, sH2, 8, lane, wave);           // H2 <- dz2 @ W2^T
    __syncthreads();
    for (int e = tid; e < 16 * HP; e += 128) {      // dx2; dz1
      int j = e & 127;
      float dx2 = sW3c[j] + sH2[e];
      sX[e] = dx2;
      sGZ[e] = dx2 * (1.f - sH1[e] * sH1[e]);
    }
    __syncthreads();
    gemm_bT(sGZ, W1, sH1, 8, lane, wave);           // H1 <- dz1 @ W1^T
    __syncthreads();
    for (int e = tid; e < 16 * HP; e += 128) {      // dx1; dz0
      float dx1 = sX[e] + sH1[e];
      sX[e] = dx1;
      sGZ[e] = dx1 * (1.f - sH0[e] * sH0[e]);
    }
    __syncthreads();
    gemm_bT(sGZ, W0, sH1, 4, lane, wave);           // H1[:, 0:64] <- dz0 @ W0^T
    __syncthreads();

    // ---- masked writes ----
    const int tt = (t == 0) ? 1 : 8;
    for (int e = tid; e < 16 * FEATP; e += 128) {
      int a = e >> 6, j = e & 63;
      if (sTy[a] == tt) {
        float v = 0.f;
        if (j < FEATN)  // W0 path + doubling-skip path
          v = sH1[a * HP + j] + sX[a * HP + j] + sX[a * HP + j + FEATN];
        dE[(base + a) * FEATP + j] = v;
      }
    }
    if (tid < 16 && sTy[tid] == tt) {
      const int g = base + tid;
      out[4 + g] = sEi[tid];                 // Ei
      atomicAdd(&out[g >> 10], sEi[tid]);    // Etot[b]
    }
    __syncthreads();
  }
}

// ---------------- force: dE . dfeat contraction + scatter ----------------
__global__ void force_kernel(const int* __restrict__ itype,
                             const int* __restrict__ nbr,
                             const float* __restrict__ dR,
                             const float* __restrict__ dE,
                             float* __restrict__ out) {
  __shared__ float sdE[FEATP];
  const int g = blockIdx.x;
  const int b = g >> 10;
  const int nloc = g & 1023;
  const int tid = threadIdx.x;
  if (tid < FEATP) sdE[tid] = dE[g * FEATP + tid];
  __syncthreads();
  float* F = out + 4 + NATOM;  // Force region [B][N][3]
  if (tid < MM) {
    const int idx = g * MM + tid;
    const int j = nbr[idx];
    const float r = dR[idx * 4 + 0];
    const int nt = (j == 0) ? 0 : itype[b * NN + (j - 1)];
    const int off = (nt == 1) ? 0 : (nt == 8) ? NBF : -1;
    if (off >= 0 && r > 0.f && r < RMAXF) {
      const float win  = 0.5f * cosf(PI_F * r) + 0.5f;
      const float dwin = -0.5f * PI_F * sinf(PI_F * r);
      float s = 0.f;
      #pragma unroll
      for (int i = 0; i < NBF; ++i) {
        float d = r - (0.5f + (float)i * (5.5f / 24.f));
        float e = expf(-d * d);
        s += sdE[off + i] * (e * (-2.f * d * win + dwin));
      }
      const float inv = 1.f / r;
      const float fx = s * dR[idx * 4 + 1] * inv;
      const float fy = s * dR[idx * 4 + 2] * inv;
      const float fz = s * dR[idx * 4 + 3] * inv;
      atomicAdd(&F[(b * NN + nloc) * 3 + 0], -fx);
      atomicAdd(&F[(b * NN + nloc) * 3 + 1], -fy);
      atomicAdd(&F[(b * NN + nloc) * 3 + 2], -fz);
      if (j > 0) {
        atomicAdd(&F[(b * NN + (j - 1)) * 3 + 0], fx);
        atomicAdd(&F[(b * NN + (j - 1)) * 3 + 1], fy);
        atomicAdd(&F[(b * NN + (j - 1)) * 3 + 2], fz);
      }
    }
  }
}

// ---------------- launch ----------------
extern "C" void kernel_launch(void* const* d_in, const int* in_sizes, int n_in,
                              void* d_out, int out_size, void* d_ws, size_t ws_size,
                              hipStream_t stream) {
  (void)in_sizes; (void)n_in; (void)out_size; (void)ws_size;
  const int*   itype = (const int*)  d_in[0];
  const int*   nbr   = (const int*)  d_in[1];
  const float* dR    = (const float*)d_in[2];
  const float* W0    = (const float*)d_in[3];
  const float* b0    = (const float*)d_in[4];
  const float* W1    = (const float*)d_in[5];
  const float* b1    = (const float*)d_in[6];
  const float* W2    = (const float*)d_in[7];
  const float* b2    = (const float*)d_in[8];
  const float* W3    = (const float*)d_in[9];
  const float* b3    = (const float*)d_in[10];
  float* out = (float*)d_out;

  float* ws = (float*)d_ws;
  float* feat = ws;                 ws += NATOM * FEATP;   // 262144
  float* dE   = ws;                 ws += NATOM * FEATP;   // 262144
  float* W0p  = ws;                 ws += 2 * FEATP * HP;  // 16384
  float* W1p  = ws;                 ws += 2 * HP * HP;     // 32768
  float* W2p  = ws;                 ws += 2 * HP * HP;     // 32768
  float* W3p  = ws;                 ws += 2 * HP * 16;     // 4096
  float* b0p  = ws;                 ws += 2 * HP;
  float* b1p  = ws;                 ws += 2 * HP;
  float* b2p  = ws;                 ws += 2 * HP;
  float* b3p  = ws;                 ws += 2 * 16;

  auto pad = [&](const float* s, float* d, int T, int K, int N, int KP, int NP) {
    int tot = T * KP * NP;
    pad_w<<<(tot + 255) / 256, 256, 0, stream>>>(s, d, T, K, N, KP, NP);
  };
  pad(W0, W0p, 2, FEATN, 100, FEATP, HP);
  pad(W1, W1p, 2, 100, 100, HP, HP);
  pad(W2, W2p, 2, 100, 100, HP, HP);
  pad(W3, W3p, 2, 100, 1, HP, 16);
  pad(b0, b0p, 2, 1, 100, 1, HP);
  pad(b1, b1p, 2, 1, 100, 1, HP);
  pad(b2, b2p, 2, 1, 100, 1, HP);
  pad(b3, b3p, 2, 1, 1, 1, 16);

  feat_kernel<<<NATOM, 128, 0, stream>>>(itype, nbr, dR, feat);
  zero_out<<<(4 + NATOM + NATOM * 3 + 255) / 256, 256, 0, stream>>>(
      out, 4 + NATOM + NATOM * 3);
  mlp_kernel<<<NATOM / 16, 128, 0, stream>>>(itype, feat, W0p, W1p, W2p, W3p,
                                             b0p, b1p, b2p, b3p, dE, out);
  force_kernel<<<NATOM, 128, 0, stream>>>(itype, nbr, dR, dE, out);
}